// Interpolator_22067541967314
// MI455X (gfx1250) — compile-verified
//
#include <hip/hip_runtime.h>

// Problem constants (match reference)
#define N_IN_C  66049
#define N_OUT_C 263169
#define B_C     128
#define K_C     4

// ---------------------------------------------------------------------------
// Pass 1: transpose x (B=128 rows x N_IN cols) -> xT (N_IN rows x 128 cols)
// so that for a fixed gather index r, xT[r*128 + b] is one contiguous,
// 16B-aligned 512B row. All index math kept in 32 bits (max idx 8.45M).
// ---------------------------------------------------------------------------
__global__ void __launch_bounds__(256)
transpose_kernel(const float* __restrict__ x, float* __restrict__ xT) {
    __shared__ float tile[32][33];          // +1 pad: conflict-free transpose
    const int tx = threadIdx.x;             // 0..31
    const int ty = threadIdx.y;             // 0..7
    const int n0 = blockIdx.x * 32;
    const int b0 = blockIdx.y * 32;

    #pragma unroll
    for (int i = 0; i < 4; ++i) {
        const int b = b0 + ty + i * 8;      // always < 128
        const int n = n0 + tx;
        if (n < N_IN_C) {
            // x is read exactly once -> NT load, keep it out of L2.
            tile[ty + i * 8][tx] =
                __builtin_nontemporal_load(&x[b * N_IN_C + n]);
        }
    }
    __syncthreads();
    #pragma unroll
    for (int i = 0; i < 4; ++i) {
        const int n = n0 + ty + i * 8;
        const int b = b0 + tx;
        if (n < N_IN_C) {
            // Regular (RT) store: xT must stay hot in the 192MB L2 for pass 2.
            xT[n * B_C + b] = tile[tx][ty + i * 8];
        }
    }
}

// ---------------------------------------------------------------------------
// Pass 2: gather-dot. 256 threads = 8 wave32s per block; block owns a tile of
// 32 output columns j x all 128 batches b.
//   compute: wave w owns j_local = 4w..4w+3; lane l owns batch quad b=4l..4l+3.
//            Every gather is ONE wave-wide float4 load of a full 512B xT row.
//   output:  results transposed through LDS so the 134MB write-once output is
//            lane-coalesced along j with non-temporal (th:TH_STORE_NT) stores,
//            preserving xT's L2 residency.
// ---------------------------------------------------------------------------
__global__ void __launch_bounds__(256)
gather_dot_kernel(const float* __restrict__ xT,
                  const int*   __restrict__ rows,
                  const float* __restrict__ vals,
                  float*       __restrict__ out) {
    __shared__ int   srow[128];             // 32 j * 4 k
    __shared__ float sval[128];
    __shared__ float stile[32 * 132];       // [j_local][b], row stride 132 (16B aligned)

    const int t  = threadIdx.x;             // 0..255
    const int w  = t >> 5;                  // wave 0..7
    const int l  = t & 31;                  // lane 0..31
    const int j0 = blockIdx.x * 32;

    // Stage rows/vals tile into LDS (coalesced; read-once -> NT loads).
    if (t < 128) {
        int gidx = j0 * 4 + t;
        if (gidx > N_OUT_C * K_C - 1) gidx = N_OUT_C * K_C - 1;  // clamp tail
        srow[t] = __builtin_nontemporal_load(&rows[gidx]);
        sval[t] = __builtin_nontemporal_load(&vals[gidx]);
    }
    __syncthreads();

    const float4* __restrict__ xT4 = (const float4*)xT;   // 32 float4 per row

    // Each wave: 4 j's, 4 k's each -> 16 wave-wide b128 gathers, all coalesced.
    #pragma unroll
    for (int q = 0; q < 4; ++q) {
        const int jl = w * 4 + q;
        const int   r0 = srow[jl * 4 + 0], r1 = srow[jl * 4 + 1];
        const int   r2 = srow[jl * 4 + 2], r3 = srow[jl * 4 + 3];
        const float v0 = sval[jl * 4 + 0], v1 = sval[jl * 4 + 1];
        const float v2 = sval[jl * 4 + 2], v3 = sval[jl * 4 + 3];

        const float4 a0 = xT4[r0 * 32 + l];
        const float4 a1 = xT4[r1 * 32 + l];
        const float4 a2 = xT4[r2 * 32 + l];
        const float4 a3 = xT4[r3 * 32 + l];

        float4 acc;
        acc.x = a0.x * v0; acc.y = a0.y * v0; acc.z = a0.z * v0; acc.w = a0.w * v0;
        acc.x = fmaf(a1.x, v1, acc.x); acc.y = fmaf(a1.y, v1, acc.y);
        acc.z = fmaf(a1.z, v1, acc.z); acc.w = fmaf(a1.w, v1, acc.w);
        acc.x = fmaf(a2.x, v2, acc.x); acc.y = fmaf(a2.y, v2, acc.y);
        acc.z = fmaf(a2.z, v2, acc.z); acc.w = fmaf(a2.w, v2, acc.w);
        acc.x = fmaf(a3.x, v3, acc.x); acc.y = fmaf(a3.y, v3, acc.y);
        acc.z = fmaf(a3.z, v3, acc.z); acc.w = fmaf(a3.w, v3, acc.w);

        // stile[jl][b], lane l covers b = 4l..4l+3; word offset multiple of 4.
        *(float4*)&stile[jl * 132 + 4 * l] = acc;
    }
    __syncthreads();

    // Write out: lane sweeps j (contiguous), loop+wave sweep b.
    const int j = j0 + l;
    if (j < N_OUT_C) {
        #pragma unroll
        for (int i = 0; i < 16; ++i) {
            const int b = i * 8 + w;
            __builtin_nontemporal_store(stile[l * 132 + b],
                                        &out[(size_t)b * N_OUT_C + j]);
        }
    }
}

// ---------------------------------------------------------------------------
// Fallback (workspace too small for the 33.8MB transpose): direct gather.
// ---------------------------------------------------------------------------
__global__ void __launch_bounds__(256)
direct_kernel(const float* __restrict__ x,
              const int*   __restrict__ rows,
              const float* __restrict__ vals,
              float*       __restrict__ out) {
    const int j = blockIdx.x * blockDim.x + threadIdx.x;
    if (j >= N_OUT_C) return;
    const int4   r = ((const int4*)rows)[j];
    const float4 v = ((const float4*)vals)[j];
    for (int b = 0; b < B_C; ++b) {
        const float* xb = x + b * N_IN_C;
        float a =      xb[r.x] * v.x;
        a = fmaf(xb[r.y], v.y, a);
        a = fmaf(xb[r.z], v.z, a);
        a = fmaf(xb[r.w], v.w, a);
        __builtin_nontemporal_store(a, &out[(size_t)b * N_OUT_C + j]);
    }
}

extern "C" void kernel_launch(void* const* d_in, const int* in_sizes, int n_in,
                              void* d_out, int out_size, void* d_ws, size_t ws_size,
                              hipStream_t stream) {
    const float* x    = (const float*)d_in[0];
    const int*   rows = (const int*)  d_in[1];
    const float* vals = (const float*)d_in[2];
    float*       out  = (float*)d_out;

    const size_t needed = (size_t)N_IN_C * B_C * sizeof(float);  // ~33.8 MB
    if (ws_size >= needed) {
        float* xT = (float*)d_ws;
        dim3 tb(32, 8);
        dim3 tg((N_IN_C + 31) / 32, B_C / 32);
        transpose_kernel<<<tg, tb, 0, stream>>>(x, xT);

        const int nblk = (N_OUT_C + 31) / 32;   // 8225 tiles
        gather_dot_kernel<<<nblk, 256, 0, stream>>>(xT, rows, vals, out);
    } else {
        direct_kernel<<<(N_OUT_C + 255) / 256, 256, 0, stream>>>(x, rows, vals, out);
    }
}